// GraphLSTMBlock_70282844831746
// MI455X (gfx1250) — compile-verified
//
#include <hip/hip_runtime.h>
#include <math.h>

typedef __attribute__((ext_vector_type(2))) float v2f;
typedef __attribute__((ext_vector_type(8))) float v8f;

constexpr int N = 4096;   // nodes
constexpr int D = 256;    // input dim
constexpr int H = 256;    // hidden dim
constexpr int NBLK = 64;  // phase-1 reduction blocks
constexpr int ROWS = N / NBLK;

__device__ __forceinline__ float sigmoidf_fast(float x) {
    return 1.0f / (1.0f + __expf(-x));
}

// ---------------------------------------------------------------------------
// init: hidden = h0, cell = c0
// ---------------------------------------------------------------------------
__global__ void __launch_bounds__(256)
k_init(const float* __restrict__ h0, const float* __restrict__ c0,
       float* __restrict__ hidden, float* __restrict__ cell) {
    size_t idx = (size_t)blockIdx.x * 256 + threadIdx.x;
    if (idx < (size_t)N * H) {
        hidden[idx] = h0[idx];
        cell[idx]   = c0[idx];
    }
}

// ---------------------------------------------------------------------------
// HW = h0 @ Wn   (4096x256 @ 256x256, f32) via V_WMMA_F32_16X16X4_F32.
// One wave computes a 16x16 tile; K-loop of 64 WMMA ops.
// A (16x4): lanes 0-15 = rows M; VGPR0 = K{0 | 2}, VGPR1 = K{1 | 3}.
// B (4x16): lanes 0-15 = cols N; VGPR0 = K{0 | 2}, VGPR1 = K{1 | 3}.
// C/D (16x16): VGPR r = row r (lanes 0-15) / row r+8 (lanes 16-31).
// ---------------------------------------------------------------------------
__global__ void __launch_bounds__(128)
k_gemm_hw_wmma(const float* __restrict__ A,   // [N,H] = h0
               const float* __restrict__ Wn,  // [H,H]
               float* __restrict__ HW) {      // [N,H]
    const int lane  = threadIdx.x & 31;
    const int wave  = threadIdx.x >> 5;
    const int tile  = blockIdx.x * 4 + wave;      // (N/16)*(H/16) = 4096 tiles
    const int tilesN = H / 16;
    const int tm = (tile / tilesN) * 16;
    const int tn = (tile % tilesN) * 16;
    const int half = lane >> 4;                   // 0: lanes 0-15, 1: lanes 16-31
    const int l    = lane & 15;

    v8f acc = {};
    for (int k = 0; k < H; k += 4) {
        const int kb = k + 2 * half;
        v2f a, b;
        a.x = A[(size_t)(tm + l) * H + kb];
        a.y = A[(size_t)(tm + l) * H + kb + 1];
        b.x = Wn[(size_t)kb * H + (tn + l)];
        b.y = Wn[(size_t)(kb + 1) * H + (tn + l)];
        acc = __builtin_amdgcn_wmma_f32_16x16x4_f32(
            /*neg_a=*/false, a, /*neg_b=*/false, b,
            /*c_mod=*/(short)0, acc, /*reuse_a=*/false, /*reuse_b=*/false);
    }
#pragma unroll
    for (int r = 0; r < 8; ++r) {
        const int row = tm + r + 8 * half;
        HW[(size_t)row * H + (tn + l)] = acc[r];
    }
}

// ---------------------------------------------------------------------------
// phase0 (1 block, 256 thr): per-step small vec-mats that depend only on the
// state at step start:  g = cat(inp,h_i)@Wg + bg,  fs = inp@Ws + bs,
// hWn = h_i@Wn.
// ---------------------------------------------------------------------------
__global__ void __launch_bounds__(256)
k_phase0(const float* __restrict__ inputs, const int* __restrict__ seq,
         const float* __restrict__ hidden,
         const float* __restrict__ Wg, const float* __restrict__ bg,
         const float* __restrict__ Ws, const float* __restrict__ bs,
         const float* __restrict__ Wn,
         float* __restrict__ fs, float* __restrict__ g, float* __restrict__ hWn,
         int t) {
    const int h = threadIdx.x;
    const int i = seq[t];
    const float* __restrict__ inp = inputs + (size_t)i * D;
    const float* __restrict__ hi  = hidden + (size_t)i * H;

    float accg = bg[h];
    for (int d = 0; d < D; ++d) accg = fmaf(inp[d], Wg[(size_t)d * H + h], accg);
    for (int k = 0; k < H; ++k) accg = fmaf(hi[k], Wg[(size_t)(D + k) * H + h], accg);

    float accf = bs[h];
    for (int d = 0; d < D; ++d) accf = fmaf(inp[d], Ws[(size_t)d * H + h], accf);

    float acch = 0.0f;
    for (int k = 0; k < H; ++k) acch = fmaf(hi[k], Wn[(size_t)k * H + h], acch);

    g[h]   = accg;
    fs[h]  = accf;
    hWn[h] = acch;
}

// ---------------------------------------------------------------------------
// phase1 (NBLK blocks x 256 thr): deterministic partial column reductions
//   Pnei[blk,h] = sum_{j in slice} w_j * hidden[j,h]
//   Pv  [blk,h] = sum_{j in slice} w_j * sigmoid(fs[h] + HW[j,h])
// Reads are fully coalesced (lane == column h).
// ---------------------------------------------------------------------------
__global__ void __launch_bounds__(256)
k_phase1(const float* __restrict__ neighbour, const int* __restrict__ seq,
         const float* __restrict__ hidden, const float* __restrict__ HW,
         const float* __restrict__ fs,
         float* __restrict__ Pnei, float* __restrict__ Pv, int t) {
    const int h   = threadIdx.x;
    const int blk = blockIdx.x;
    const int i   = seq[t];
    const float f = fs[h];
    const int jbase = blk * ROWS;
    const float* __restrict__ wrow = neighbour + (size_t)i * N + jbase;

    float sn = 0.0f, sv = 0.0f;
    for (int r = 0; r < ROWS; ++r) {
        const size_t off = (size_t)(jbase + r) * H + h;
        if (r + 4 < ROWS) {
            __builtin_prefetch(&hidden[off + 4 * (size_t)H], 0, 1);
            __builtin_prefetch(&HW[off + 4 * (size_t)H], 0, 1);
        }
        const float w = wrow[r];
        sn = fmaf(w, hidden[off], sn);
        sv = fmaf(w, sigmoidf_fast(f + HW[off]), sv);
    }
    Pnei[(size_t)blk * H + h] = sn;
    Pv[(size_t)blk * H + h]   = sv;
}

// ---------------------------------------------------------------------------
// phase2 (1 block, 256 thr): reduce partials, nonlinearities, state update,
// and incremental HW[i] = curHidden @ Wn.
// ---------------------------------------------------------------------------
__global__ void __launch_bounds__(256)
k_phase2(const int* __restrict__ seq, const float* __restrict__ numNei,
         const float* __restrict__ Pnei, const float* __restrict__ Pv,
         const float* __restrict__ fsArr, const float* __restrict__ gArr,
         const float* __restrict__ hWnArr, const float* __restrict__ Wn,
         float* __restrict__ hidden, float* __restrict__ cell,
         float* __restrict__ HW, int t) {
    __shared__ float s_neiH[H];
    __shared__ float s_curH[H];
    const int h = threadIdx.x;
    const int i = seq[t];
    const float n = numNei[i];

    float sn = 0.0f, sv = 0.0f;
    for (int b = 0; b < NBLK; ++b) {
        sn += Pnei[(size_t)b * H + h];
        sv += Pv[(size_t)b * H + h];
    }
    s_neiH[h] = sn / n;
    __syncthreads();

    float acc = 0.0f;
    for (int k = 0; k < H; ++k) acc = fmaf(s_neiH[k], Wn[(size_t)k * H + h], acc);

    const float pre         = gArr[h] + acc;
    const float inputState  = sigmoidf_fast(pre);   // == outputState in original
    const float helpCell    = tanhf(pre);
    const float forgetState = sigmoidf_fast(fsArr[h] + hWnArr[h]);

    const size_t idx = (size_t)i * H + h;
    const float ci = cell[idx];
    // nei_row @ (neiForget * c_i) == (nei_row @ neiForget) * c_i == sv * ci
    const float curCell   = (sv / n) * ci + forgetState * ci + inputState * helpCell;
    const float curHidden = tanhf(inputState * curCell);

    cell[idx]   = curCell;
    hidden[idx] = curHidden;
    s_curH[h]   = curHidden;
    __syncthreads();

    float accH = 0.0f;
    for (int k = 0; k < H; ++k) accH = fmaf(s_curH[k], Wn[(size_t)k * H + h], accH);
    HW[idx] = accH;   // incremental update: HW stays == hidden @ Wn
}

// ---------------------------------------------------------------------------
// out = hidden + h0
// ---------------------------------------------------------------------------
__global__ void __launch_bounds__(256)
k_finalize(const float* __restrict__ hidden, const float* __restrict__ h0,
           float* __restrict__ out) {
    size_t idx = (size_t)blockIdx.x * 256 + threadIdx.x;
    if (idx < (size_t)N * H) out[idx] = hidden[idx] + h0[idx];
}

// ---------------------------------------------------------------------------
extern "C" void kernel_launch(void* const* d_in, const int* in_sizes, int n_in,
                              void* d_out, int out_size, void* d_ws, size_t ws_size,
                              hipStream_t stream) {
    const float* inputs    = (const float*)d_in[0];
    const float* neighbour = (const float*)d_in[1];
    const float* numNei    = (const float*)d_in[2];
    const int*   sequence  = (const int*)  d_in[3];
    const float* h0        = (const float*)d_in[4];
    const float* c0        = (const float*)d_in[5];
    const float* Wg        = (const float*)d_in[6];
    // const float* bg     = (const float*)d_in[7];
    const float* Ws        = (const float*)d_in[8];
    // const float* bs     = (const float*)d_in[9];
    const float* Wn        = (const float*)d_in[10];
    const float* bg        = (const float*)d_in[7];
    const float* bs        = (const float*)d_in[9];
    float* out = (float*)d_out;
    float* ws  = (float*)d_ws;

    const int S = in_sizes[3];
    const size_t NH = (size_t)N * H;

    // workspace layout (floats)
    float* hidden = ws;                 // [N,H]
    float* cell   = hidden + NH;        // [N,H]
    float* HW     = cell + NH;          // [N,H]  == hidden @ Wn (maintained)
    float* fsBuf  = HW + NH;            // [H]
    float* gBuf   = fsBuf + H;          // [H]
    float* hWnBuf = gBuf + H;           // [H]
    float* Pnei   = hWnBuf + H;         // [NBLK,H]
    float* Pv     = Pnei + (size_t)NBLK * H; // [NBLK,H]

    const int ngrid = (int)((NH + 255) / 256);
    k_init<<<ngrid, 256, 0, stream>>>(h0, c0, hidden, cell);

    // HW = h0 @ Wn via f32 WMMA: 4096 16x16 tiles, 4 waves per block
    k_gemm_hw_wmma<<<(N / 16) * (H / 16) / 4, 128, 0, stream>>>(h0, Wn, HW);

    for (int t = 0; t < S; ++t) {
        k_phase0<<<1, 256, 0, stream>>>(inputs, sequence, hidden,
                                        Wg, bg, Ws, bs, Wn,
                                        fsBuf, gBuf, hWnBuf, t);
        k_phase1<<<NBLK, 256, 0, stream>>>(neighbour, sequence, hidden, HW,
                                           fsBuf, Pnei, Pv, t);
        k_phase2<<<1, 256, 0, stream>>>(sequence, numNei, Pnei, Pv,
                                        fsBuf, gBuf, hWnBuf, Wn,
                                        hidden, cell, HW, t);
    }

    k_finalize<<<ngrid, 256, 0, stream>>>(hidden, h0, out);
}